// GraphormerBlock_76888504533068
// MI455X (gfx1250) — compile-verified
//
#include <hip/hip_runtime.h>
#include <hip/hip_bf16.h>
#include <math.h>

typedef __attribute__((ext_vector_type(2))) float v2f;
typedef __attribute__((ext_vector_type(8))) float v8f;
typedef __attribute__((ext_vector_type(4))) unsigned int u32x4;
typedef __attribute__((ext_vector_type(8))) int i32x8;
typedef __attribute__((ext_vector_type(4))) int i32x4;

__device__ __forceinline__ v8f wmma_f32(v2f a, v2f b, v8f c) {
  // D = A(16x4,f32) * B(4x16,f32) + C(16x16,f32); wave32
  return __builtin_amdgcn_wmma_f32_16x16x4_f32(false, a, false, b, (short)0, c,
                                               false, false);
}

__device__ __forceinline__ v8f zero8() {
  v8f z = {0.f, 0.f, 0.f, 0.f, 0.f, 0.f, 0.f, 0.f};
  return z;
}

#define GBM 128
#define GBN 64
#define GKS 16
#define GSTR (GKS + 2) // LDS row stride (floats); 18 -> conflict-free b64 reads

#if __has_builtin(__builtin_amdgcn_tensor_load_to_lds) && \
    __has_builtin(__builtin_amdgcn_s_wait_tensorcnt)
#define USE_TDM 1
#else
#define USE_TDM 0
#endif

#if USE_TDM
// LDS byte offset = low 32 bits of the generic pointer (LDS aperture base has
// zero low bits).
__device__ __forceinline__ unsigned lds_off(const void* p) {
  return (unsigned)(unsigned long long)(size_t)p;
}

// TDM 2D tile load: tile_w=16 f32 elems/row, tile_h rows, row stride = K.
// D# pad: after every 16 DWORDs insert 2 DWORDs -> LDS row stride 18 floats,
// matching the GSTR layout the WMMA fragment reads expect.
__device__ __forceinline__ void tdm_load_tile(unsigned lds_byte,
                                              const float* g, int tile_h,
                                              int row_stride) {
  const unsigned long long ga = (unsigned long long)(size_t)g;
  u32x4 g0;
  g0[0] = 1u; // count=1 (valid), user mode, no gather
  g0[1] = lds_byte;
  g0[2] = (unsigned)ga;
  g0[3] = (unsigned)((ga >> 32) & 0x1FFFFFFull) | (2u << 30); // type=2 (image)
  i32x8 g1;
  // data_size=4B(2), pad_enable, pad_interval=16DW(3), pad_amount=2DW(1)
  g1[0] = (int)((2u << 16) | (1u << 20) | (3u << 22) | (1u << 25));
  g1[1] = (int)(((unsigned)row_stride & 0xFFFFu) << 16); // tensor_dim0 lo16
  g1[2] = (int)((((unsigned)row_stride >> 16) & 0xFFFFu) |
                (8192u << 16));           // tensor_dim0 hi16 | tensor_dim1 lo16
  g1[3] = (int)((unsigned)GKS << 16);     // tensor_dim1 hi16=0 | tile_dim0=16
  g1[4] = tile_h;                         // tile_dim1 | tile_dim2=0
  g1[5] = row_stride;                     // tensor_dim0_stride lo32
  g1[6] = 0;                              // stride hi16 | dim1_stride lo16
  g1[7] = 0;
  i32x4 gz = {0, 0, 0, 0};
#if __clang_major__ >= 23
  i32x8 gz8 = {0, 0, 0, 0, 0, 0, 0, 0};
  __builtin_amdgcn_tensor_load_to_lds(g0, g1, gz, gz, gz8, 0);
#else
  __builtin_amdgcn_tensor_load_to_lds(g0, g1, gz, gz, 0);
#endif
}
#endif // USE_TDM

// ---------------------------------------------------------------------------
// Generic tiled GEMM: out = A(MxK) * W(NxK)^T  (torch Linear convention)
// Block tile 128x64, 8 waves in 4x2 grid, each wave 32x32 (4 WMMA C tiles).
// Tiles staged in LDS by the Tensor Data Mover, double buffered so the DMA of
// step i+1 overlaps the 16 WMMAs of step i.
// mode 0: write permuted (B,H,N,dk) QKV layout, no bias
// mode 1: row-major + optional bias
// mode 2: row-major + bias + exact GELU
// ---------------------------------------------------------------------------
__global__ __launch_bounds__(256) void gemm_kernel(
    const float* __restrict__ A, const float* __restrict__ W,
    float* __restrict__ out, const float* __restrict__ bias, int M, int N,
    int K, int mode) {
  __shared__ float As[2][GBM * GSTR]; // 2 x 9216 B
  __shared__ float Ws[2][GBN * GSTR]; // 2 x 4608 B

  const int tid = threadIdx.x;
  const int lane = tid & 31;
  const int wv = tid >> 5;  // 0..7
  const int wm = wv >> 1;   // 0..3  (M direction)
  const int wn = wv & 1;    // 0..1  (N direction)
  const int hf = lane >> 4; // lane half
  const int l15 = lane & 15;
  const int m0 = blockIdx.x * GBM;
  const int n0 = blockIdx.y * GBN;

  v8f acc[2][2];
  acc[0][0] = zero8(); acc[0][1] = zero8();
  acc[1][0] = zero8(); acc[1][1] = zero8();

  const int nsteps = K / GKS;

#if USE_TDM
  if (wv == 0) { // one wave drives the TDM; EXEC is ignored by tensor ops
    tdm_load_tile(lds_off(&As[0][0]), A + (size_t)m0 * K, GBM, K);
    tdm_load_tile(lds_off(&Ws[0][0]), W + (size_t)n0 * K, GBN, K);
  }
  for (int i = 0; i < nsteps; ++i) {
    const int cur = i & 1;
    if (wv == 0) {
      if (i + 1 < nsteps) {
        const int k1 = (i + 1) * GKS;
        // Prefetch next pair into the other buffer (its readers finished at
        // the trailing barrier of iteration i-1).
        tdm_load_tile(lds_off(&As[cur ^ 1][0]), A + (size_t)m0 * K + k1, GBM, K);
        tdm_load_tile(lds_off(&Ws[cur ^ 1][0]), W + (size_t)n0 * K + k1, GBN, K);
        __builtin_amdgcn_s_wait_tensorcnt(2); // in-order: pair i now complete
      } else {
        __builtin_amdgcn_s_wait_tensorcnt(0);
      }
    }
    __syncthreads(); // publish buffer `cur`
    const float* as = &As[cur][0];
    const float* wsb = &Ws[cur][0];
#pragma unroll
    for (int kk = 0; kk < GKS; kk += 4) {
      const int ko = kk + 2 * hf; // per-half K offset (ISA fp32 A/B layout)
      v2f af0 = *(const v2f*)&as[(wm * 32 + l15) * GSTR + ko];
      v2f af1 = *(const v2f*)&as[(wm * 32 + 16 + l15) * GSTR + ko];
      v2f bf0 = *(const v2f*)&wsb[(wn * 32 + l15) * GSTR + ko];
      v2f bf1 = *(const v2f*)&wsb[(wn * 32 + 16 + l15) * GSTR + ko];
      acc[0][0] = wmma_f32(af0, bf0, acc[0][0]);
      acc[0][1] = wmma_f32(af0, bf1, acc[0][1]);
      acc[1][0] = wmma_f32(af1, bf0, acc[1][0]);
      acc[1][1] = wmma_f32(af1, bf1, acc[1][1]);
    }
    __syncthreads(); // release buffer `cur` for the TDM write in step i+2
  }
#else
  for (int i = 0; i < nsteps; ++i) {
    const int k0 = i * GKS;
    __syncthreads();
    {
      const int f0 = tid, f1 = tid + 256;
      const int r0 = f0 >> 2, kc0 = (f0 & 3) << 2;
      const int r1 = f1 >> 2, kc1 = (f1 & 3) << 2;
      const float4 t0 = *(const float4*)(A + (size_t)(m0 + r0) * K + k0 + kc0);
      const float4 t1 = *(const float4*)(A + (size_t)(m0 + r1) * K + k0 + kc1);
      float* d0 = &As[0][r0 * GSTR + kc0];
      float* d1 = &As[0][r1 * GSTR + kc1];
      d0[0] = t0.x; d0[1] = t0.y; d0[2] = t0.z; d0[3] = t0.w;
      d1[0] = t1.x; d1[1] = t1.y; d1[2] = t1.z; d1[3] = t1.w;
      const int rw = tid >> 2, kcw = (tid & 3) << 2;
      const float4 tw = *(const float4*)(W + (size_t)(n0 + rw) * K + k0 + kcw);
      float* dw = &Ws[0][rw * GSTR + kcw];
      dw[0] = tw.x; dw[1] = tw.y; dw[2] = tw.z; dw[3] = tw.w;
    }
    __syncthreads();
#pragma unroll
    for (int kk = 0; kk < GKS; kk += 4) {
      const int ko = kk + 2 * hf;
      v2f af0 = *(const v2f*)&As[0][(wm * 32 + l15) * GSTR + ko];
      v2f af1 = *(const v2f*)&As[0][(wm * 32 + 16 + l15) * GSTR + ko];
      v2f bf0 = *(const v2f*)&Ws[0][(wn * 32 + l15) * GSTR + ko];
      v2f bf1 = *(const v2f*)&Ws[0][(wn * 32 + 16 + l15) * GSTR + ko];
      acc[0][0] = wmma_f32(af0, bf0, acc[0][0]);
      acc[0][1] = wmma_f32(af0, bf1, acc[0][1]);
      acc[1][0] = wmma_f32(af1, bf0, acc[1][0]);
      acc[1][1] = wmma_f32(af1, bf1, acc[1][1]);
    }
  }
#endif

#pragma unroll
  for (int ti = 0; ti < 2; ++ti) {
#pragma unroll
    for (int tj = 0; tj < 2; ++tj) {
#pragma unroll
      for (int j = 0; j < 8; ++j) {
        const int r = m0 + wm * 32 + ti * 16 + hf * 8 + j;
        const int c = n0 + wn * 32 + tj * 16 + l15;
        float val = acc[ti][tj][j];
        if (mode == 0) {
          // (B*N, DIM) -> (B, H, N, dk)
          const int b = r >> 10, n = r & 1023;
          const int h = c >> 5, dk = c & 31;
          out[((((size_t)(b * 8 + h)) << 10) + n) * 32 + dk] = val;
        } else {
          if (bias) val += bias[c];
          if (mode == 2)
            val = 0.5f * val * (1.0f + erff(val * 0.70710678118654752f));
          out[(size_t)r * N + c] = val;
        }
      }
    }
  }
}

// ---------------------------------------------------------------------------
// Attention: one block per (b*h, 16-query tile). 4 waves split 1024 keys.
// scores (16x1024) staged in LDS (stride 1026 -> conflict-free b64 reads),
// LDS softmax, then attn*V via WMMA with cross-wave partial reduction.
// ---------------------------------------------------------------------------
#define SSTR 1026

__global__ __launch_bounds__(128) void attn_kernel(
    const float* __restrict__ qb, const float* __restrict__ kb,
    const float* __restrict__ vb, const int* __restrict__ adj,
    const float* __restrict__ dist_emb, float* __restrict__ obuf) {
  __shared__ float sc[16 * SSTR];   // 65664 B (gfx1250: 320KB LDS/WGP)
  __shared__ float red[16][8];      // 512 B
  __shared__ float pbuf[4][16][32]; // 8192 B

  const int tid = threadIdx.x;
  const int lane = tid & 31;
  const int wv = tid >> 5; // 0..3
  const int hf = lane >> 4;
  const int l15 = lane & 15;
  const int q0 = blockIdx.x * 16;
  const int bh = blockIdx.y; // b*8 + h
  const int b = bh >> 3, h = bh & 7;

  const float* q = qb + (((size_t)bh) << 10) * 32; // (1024, 32)
  const float* kp = kb + (((size_t)bh) << 10) * 32;
  const float* vp = vb + (((size_t)bh) << 10) * 32;
  const float de1 = dist_emb[h * 6 + 1]; // spd==1 (edge)
  const float de2 = dist_emb[h * 6 + 2]; // spd==2 (no edge, clip is no-op)
  const float scale = 0.17677669529663687f; // 1/sqrt(32)

  // Preload Q fragments (reused for all key tiles of this wave)
  v2f qf[8];
#pragma unroll
  for (int i = 0; i < 8; ++i)
    qf[i] = *(const v2f*)&q[(size_t)(q0 + l15) * 32 + i * 4 + 2 * hf];

  const int key0w = wv * 256;
  // Phase 1: scores = Q*K^T * scale + bias(adj)
  for (int kt = 0; kt < 16; ++kt) {
    const int key0 = key0w + kt * 16;
    v8f acc = zero8();
#pragma unroll
    for (int i = 0; i < 8; ++i) {
      v2f bf = *(const v2f*)&kp[(size_t)(key0 + l15) * 32 + i * 4 + 2 * hf];
      acc = wmma_f32(qf[i], bf, acc);
    }
    const int kidx = key0 + l15;
#pragma unroll
    for (int j = 0; j < 8; ++j) {
      const int qrow = hf * 8 + j;
      const int a =
          adj[(((size_t)b) << 20) + (((size_t)(q0 + qrow)) << 10) + kidx];
      const float bias = (a > 0) ? de1 : de2;
      sc[qrow * SSTR + kidx] = acc[j] * scale + bias;
    }
  }
  __syncthreads();

  // Phase 2: row softmax over 1024 keys (16 rows x 8 threads each)
  const int row = tid >> 3, sub = tid & 7;
  float pm = -3.4e38f;
  for (int i = 0; i < 128; ++i)
    pm = fmaxf(pm, sc[row * SSTR + sub + i * 8]);
  red[row][sub] = pm;
  __syncthreads();
  float rm = red[row][0];
#pragma unroll
  for (int i = 1; i < 8; ++i) rm = fmaxf(rm, red[row][i]);
  __syncthreads();
  float ps = 0.f;
  for (int i = 0; i < 128; ++i) {
    const int idx = row * SSTR + sub + i * 8;
    const float e = __expf(sc[idx] - rm);
    sc[idx] = e;
    ps += e;
  }
  red[row][sub] = ps;
  __syncthreads();
  float rs = 0.f;
#pragma unroll
  for (int i = 0; i < 8; ++i) rs += red[row][i];
  const float rinv = 1.0f / rs;
  for (int i = 0; i < 128; ++i) sc[row * SSTR + sub + i * 8] *= rinv;
  __syncthreads();

  // Phase 3: out(16x32) = attn(16x1024) * V(1024x32); each wave 256 keys
  v8f o0 = zero8(), o1 = zero8();
  for (int ks = 0; ks < 256; ks += 4) {
    const int kidx = key0w + ks + 2 * hf;
    v2f af = *(const v2f*)&sc[l15 * SSTR + kidx];
    const float* vr = &vp[(size_t)kidx * 32];
    v2f b0, b1;
    b0[0] = vr[l15];      b0[1] = vr[32 + l15];
    b1[0] = vr[16 + l15]; b1[1] = vr[48 + l15];
    o0 = wmma_f32(af, b0, o0);
    o1 = wmma_f32(af, b1, o1);
  }
#pragma unroll
  for (int j = 0; j < 8; ++j) {
    pbuf[wv][hf * 8 + j][l15] = o0[j];
    pbuf[wv][hf * 8 + j][16 + l15] = o1[j];
  }
  __syncthreads();
  // Cross-wave reduce + write (B, N, DIM) with heads re-concatenated
  for (int e = tid; e < 512; e += 128) {
    const int qr = e >> 5, dk = e & 31;
    const float s = pbuf[0][qr][dk] + pbuf[1][qr][dk] + pbuf[2][qr][dk] +
                    pbuf[3][qr][dk];
    obuf[((size_t)b * 1024 + q0 + qr) * 256 + h * 32 + dk] = s;
  }
}

// ---------------------------------------------------------------------------
// LayerNorm(residual + y): one wave per 256-wide row, 8 rows per block
// ---------------------------------------------------------------------------
__global__ __launch_bounds__(256) void ln_kernel(
    const float* __restrict__ xin, const float* __restrict__ res,
    const float* __restrict__ w, const float* __restrict__ bp,
    float* __restrict__ out) {
  const int wv = threadIdx.x >> 5, lane = threadIdx.x & 31;
  const int r = blockIdx.x * 8 + wv;
  const float* xr = xin + (size_t)r * 256;
  const float* rr = res + (size_t)r * 256;
  float vals[8];
  float s = 0.f;
#pragma unroll
  for (int i = 0; i < 8; ++i) {
    const int c = lane * 8 + i;
    vals[i] = xr[c] + rr[c];
    s += vals[i];
  }
#pragma unroll
  for (int off = 16; off; off >>= 1) s += __shfl_xor(s, off, 32);
  const float mean = s * (1.0f / 256.0f);
  float vsum = 0.f;
#pragma unroll
  for (int i = 0; i < 8; ++i) {
    const float d = vals[i] - mean;
    vsum += d * d;
  }
#pragma unroll
  for (int off = 16; off; off >>= 1) vsum += __shfl_xor(vsum, off, 32);
  const float rstd = rsqrtf(vsum * (1.0f / 256.0f) + 1e-5f);
  float* outr = out + (size_t)r * 256;
#pragma unroll
  for (int i = 0; i < 8; ++i) {
    const int c = lane * 8 + i;
    outr[c] = (vals[i] - mean) * rstd * w[c] + bp[c];
  }
}

// ---------------------------------------------------------------------------
extern "C" void kernel_launch(void* const* d_in, const int* in_sizes, int n_in,
                              void* d_out, int out_size, void* d_ws,
                              size_t ws_size, hipStream_t stream) {
  const float* x = (const float*)d_in[0];
  const int* adj = (const int*)d_in[1];
  const float* wq = (const float*)d_in[2];
  const float* wk = (const float*)d_in[3];
  const float* wvv = (const float*)d_in[4];
  const float* wo = (const float*)d_in[5];
  const float* dist = (const float*)d_in[6];
  const float* ln1w = (const float*)d_in[7];
  const float* ln1b = (const float*)d_in[8];
  const float* fw1 = (const float*)d_in[9];
  const float* fb1 = (const float*)d_in[10];
  const float* fw2 = (const float*)d_in[11];
  const float* fb2 = (const float*)d_in[12];
  const float* ln2w = (const float*)d_in[13];
  const float* ln2b = (const float*)d_in[14];
  float* out = (float*)d_out;

  float* ws = (float*)d_ws;
  const size_t T2M = 2u * 1024u * 1024u; // 2M floats per (B,N,DIM) buffer
  float* qbuf = ws + 0 * T2M;
  float* kbuf = ws + 1 * T2M;
  float* vbuf = ws + 2 * T2M;
  float* obuf = ws + 3 * T2M; // attention output (B,N,DIM)
  float* oproj = ws + 4 * T2M;
  float* x1 = ws + 5 * T2M;   // LN1 output, live until LN2
  float* hbuf = ws + 0 * T2M; // FFN hidden (8M floats) aliases q/k/v/obuf
  float* o2 = ws + 4 * T2M;   // FFN2 output aliases oproj

  const dim3 blk(256);
  // QKV projections -> (B,H,N,dk)
  gemm_kernel<<<dim3(64, 4), blk, 0, stream>>>(x, wq, qbuf, nullptr, 8192, 256, 256, 0);
  gemm_kernel<<<dim3(64, 4), blk, 0, stream>>>(x, wk, kbuf, nullptr, 8192, 256, 256, 0);
  gemm_kernel<<<dim3(64, 4), blk, 0, stream>>>(x, wvv, vbuf, nullptr, 8192, 256, 256, 0);
  // Biased attention + softmax + attn*V
  attn_kernel<<<dim3(64, 64), dim3(128), 0, stream>>>(qbuf, kbuf, vbuf, adj, dist, obuf);
  // Output projection
  gemm_kernel<<<dim3(64, 4), blk, 0, stream>>>(obuf, wo, oproj, nullptr, 8192, 256, 256, 1);
  // x1 = LN1(x + oproj)
  ln_kernel<<<dim3(1024), blk, 0, stream>>>(x, oproj, ln1w, ln1b, x1);
  // FFN: hbuf = GELU(x1 @ W1^T + b1)
  gemm_kernel<<<dim3(64, 16), blk, 0, stream>>>(x1, fw1, hbuf, fb1, 8192, 1024, 256, 2);
  // o2 = hbuf @ W2^T + b2
  gemm_kernel<<<dim3(64, 4), blk, 0, stream>>>(hbuf, fw2, o2, fb2, 8192, 256, 1024, 1);
  // out = LN2(x1 + o2)
  ln_kernel<<<dim3(1024), blk, 0, stream>>>(x1, o2, ln2w, ln2b, out);
}